// DynamicImagePrimalDualNN_13709535608863
// MI455X (gfx1250) — compile-verified
//
#include <hip/hip_runtime.h>
#include <cstdint>

// ---------------------------------------------------------------------------
// Chambolle-Pock primal-dual TV solver, (mb=2, Nx=160, Ny=160, Nt=32), T=128.
// Bandwidth-bound stencil iteration: fp32, wave32 == Nt so t-rolls are
// intra-wave LDS rotations; stencil lines staged via gfx1250 async
// global->LDS loads (ASYNCcnt path). Whole state (~66 MB) is L2-resident
// on MI455X (192 MB L2).
// ---------------------------------------------------------------------------

#define NXD 160
#define NYD 160
#define NTD 32
#define MBD 2
#define TSTEPS 128
#define CSTR   819200u     // channel stride = NX*NY*NT
#define N1ELEM 1638400u    // MB * CSTR
#define NLINES 51200       // MB*NX*NY
#define NBLOCK 6400        // NLINES / 8 lines-per-block
#define INV_L  0.2773500981126146f  // 1/sqrt(13)

#if __has_builtin(__builtin_amdgcn_global_load_async_to_lds_b32)
#define ASYNC_LDS 1
typedef __attribute__((address_space(1))) int gi32_t;
typedef __attribute__((address_space(3))) int li32_t;
#else
#define ASYNC_LDS 0
#endif

// Issue one 4-byte async global->LDS copy per lane (whole wave covers a
// 128-byte t-line). LDS destination per wave is private, so only
// s_wait_asynccnt is needed (no workgroup barrier).
__device__ __forceinline__ void async_line_b32(const float* g, float* l) {
#if ASYNC_LDS
  // AS1 pointer via integer reinterpret (64-bit); AS3 pointer is 32-bit and
  // the low 32 bits of a generic shared address are the LDS byte offset.
  __builtin_amdgcn_global_load_async_to_lds_b32(
      (gi32_t*)(uintptr_t)g, (li32_t*)(uint32_t)(uintptr_t)l, 0, 0);
#else
  *l = *g;
#endif
}

__device__ __forceinline__ void wait_async_lds() {
#if ASYNC_LDS
#if __has_builtin(__builtin_amdgcn_s_wait_asynccnt)
  __builtin_amdgcn_s_wait_asynccnt(0);
#else
  asm volatile("s_wait_asynccnt 0x0" ::: "memory");
#endif
#endif
  asm volatile("" ::: "memory");  // keep LDS reads after the wait
}

__device__ __forceinline__ float dsigmoid(float v) {
  return 1.0f / (1.0f + __expf(-v));
}

// ---------------------------------------------------------------------------
// init: p = x, q = 0, x0 = x, xbar = x
// ---------------------------------------------------------------------------
__global__ void pd_init(const float* __restrict__ x,
                        float* __restrict__ p, float* __restrict__ q,
                        float* __restrict__ x0, float* __restrict__ xbar) {
  uint32_t idx = blockIdx.x * 256u + threadIdx.x;   // N1ELEM == 6400*256 exactly
  float xv = x[idx];
  p[idx] = xv;
  x0[idx] = xv;
  xbar[idx] = xv;
  uint32_t b = idx / CSTR;
  uint32_t ip = idx - b * CSTR;
  size_t qb = (size_t)(b * 3u) * CSTR + ip;
  q[qb] = 0.0f;
  q[qb + CSTR] = 0.0f;
  q[qb + 2u * CSTR] = 0.0f;
}

// ---------------------------------------------------------------------------
// Phase A: p = (p + sig*(xbar - x)) / (1+sig)
//          q = clip( q + sig * grad_G(xbar), lambda )
// One wave per (b,i,j) line of 32 t-values. Stage xbar lines (self, i+1, j+1)
// into LDS via async loads; t+1 roll = LDS lane rotation.
// ---------------------------------------------------------------------------
__global__ void pd_phaseA(const float* __restrict__ x,
                          const float* __restrict__ lam,
                          const float* __restrict__ sigma_p,
                          float* __restrict__ p, float* __restrict__ q,
                          const float* __restrict__ xbar) {
  __shared__ float sxb[8][3][NTD];
  const int lane = threadIdx.x & 31;
  const int w = threadIdx.x >> 5;
  const int line = blockIdx.x * 8 + w;
  const int b = line / (NXD * NYD);
  const int r = line - b * NXD * NYD;
  const int i = r / NYD;
  const int j = r - i * NYD;
  const int ipn = (i + 1 == NXD) ? 0 : i + 1;
  const int jpn = (j + 1 == NYD) ? 0 : j + 1;

  const size_t inplane    = ((size_t)i * NYD + j) * NTD + lane;
  const size_t inplane_ip = ((size_t)ipn * NYD + j) * NTD + lane;
  const size_t inplane_jp = ((size_t)i * NYD + jpn) * NTD + lane;
  const size_t g1 = (size_t)b * CSTR;  // base of 1-channel arrays for batch b

  async_line_b32(xbar + g1 + inplane,    &sxb[w][0][lane]);
  async_line_b32(xbar + g1 + inplane_ip, &sxb[w][1][lane]);
  async_line_b32(xbar + g1 + inplane_jp, &sxb[w][2][lane]);

  const float sig = dsigmoid(sigma_p[0]) * INV_L;
  const float xn = x[g1 + inplane];
  const float pold = p[g1 + inplane];

  wait_async_lds();
  const float xb   = sxb[w][0][lane];
  const float xbtp = sxb[w][0][(lane + 1) & 31];  // roll(-1, t)
  const float xbip = sxb[w][1][lane];             // roll(-1, x)
  const float xbjp = sxb[w][2][lane];             // roll(-1, y)

  p[g1 + inplane] = (pold + sig * (xb - xn)) / (1.0f + sig);

  float d0 = xbip - xb;
  float d1 = xbjp - xb;
  float d2 = xbtp - xb;
  const size_t q3 = (size_t)(b * 3) * CSTR + inplane;

  {
    float qv = q[q3] + sig * d0;
    float lv = lam[q3];
    q[q3] = lv * qv / fmaxf(lv, fabsf(qv));
  }
  {
    float qv = q[q3 + CSTR] + sig * d1;
    float lv = lam[q3 + CSTR];
    q[q3 + CSTR] = lv * qv / fmaxf(lv, fabsf(qv));
  }
  {
    float qv = q[q3 + 2u * CSTR] + sig * d2;
    float lv = lam[q3 + 2u * CSTR];
    q[q3 + 2u * CSTR] = lv * qv / fmaxf(lv, fabsf(qv));
  }
}

// ---------------------------------------------------------------------------
// Phase B: div = grad_GH(q);  x1 = x0 - ta*p - ta*div
//          xbar = x1 + th*(x1 - x0);  x0 <- x1
// Stage the 5 needed q lines (qx@i, qx@i-1, qy@j, qy@j-1, qt@self) via async
// loads; t-1 roll = LDS lane rotation. On the last step also write d_out.
// ---------------------------------------------------------------------------
__global__ void pd_phaseB(const float* __restrict__ tau_p,
                          const float* __restrict__ theta_p,
                          const float* __restrict__ p,
                          const float* __restrict__ q,
                          float* __restrict__ x0, float* __restrict__ xbar,
                          float* __restrict__ out /* nullptr except last step */) {
  __shared__ float sq[8][5][NTD];
  const int lane = threadIdx.x & 31;
  const int w = threadIdx.x >> 5;
  const int line = blockIdx.x * 8 + w;
  const int b = line / (NXD * NYD);
  const int r = line - b * NXD * NYD;
  const int i = r / NYD;
  const int j = r - i * NYD;
  const int imn = (i == 0) ? NXD - 1 : i - 1;
  const int jmn = (j == 0) ? NYD - 1 : j - 1;

  const size_t inplane    = ((size_t)i * NYD + j) * NTD + lane;
  const size_t inplane_im = ((size_t)imn * NYD + j) * NTD + lane;
  const size_t inplane_jm = ((size_t)i * NYD + jmn) * NTD + lane;
  const size_t qb = (size_t)(b * 3) * CSTR;

  async_line_b32(q + qb + inplane,               &sq[w][0][lane]);  // qx(i,j)
  async_line_b32(q + qb + inplane_im,            &sq[w][1][lane]);  // qx(i-1,j)
  async_line_b32(q + qb + CSTR + inplane,        &sq[w][2][lane]);  // qy(i,j)
  async_line_b32(q + qb + CSTR + inplane_jm,     &sq[w][3][lane]);  // qy(i,j-1)
  async_line_b32(q + qb + 2u * CSTR + inplane,   &sq[w][4][lane]);  // qt(i,j)

  const float ta = dsigmoid(tau_p[0]) * INV_L;
  const float th = dsigmoid(theta_p[0]);
  const size_t g1 = (size_t)b * CSTR + inplane;
  const float pv = p[g1];
  const float x0v = x0[g1];

  wait_async_lds();
  const float qx  = sq[w][0][lane];
  const float qxm = sq[w][1][lane];
  const float qy  = sq[w][2][lane];
  const float qym = sq[w][3][lane];
  const float qt  = sq[w][4][lane];
  const float qtm = sq[w][4][(lane + 31) & 31];  // roll(+1, t)

  const float divq = (qxm - qx) + (qym - qy) + (qtm - qt);
  const float x1 = x0v - ta * pv - ta * divq;

  x0[g1]   = x1;
  xbar[g1] = x1 + th * (x1 - x0v);
  if (out) out[g1] = x1;
}

// ---------------------------------------------------------------------------
extern "C" void kernel_launch(void* const* d_in, const int* in_sizes, int n_in,
                              void* d_out, int out_size, void* d_ws, size_t ws_size,
                              hipStream_t stream) {
  (void)in_sizes; (void)n_in; (void)out_size; (void)ws_size;
  const float* x     = (const float*)d_in[0];
  const float* lam   = (const float*)d_in[1];
  const float* tau   = (const float*)d_in[2];
  const float* sigma = (const float*)d_in[3];
  const float* theta = (const float*)d_in[4];
  float* out = (float*)d_out;

  float* ws   = (float*)d_ws;
  float* p    = ws;                 // N1ELEM
  float* q    = ws + N1ELEM;        // 3*N1ELEM
  float* x0   = ws + 4u * N1ELEM;   // N1ELEM
  float* xbar = ws + 5u * N1ELEM;   // N1ELEM

  pd_init<<<NBLOCK, 256, 0, stream>>>(x, p, q, x0, xbar);
  for (int s = 0; s < TSTEPS; ++s) {
    pd_phaseA<<<NBLOCK, 256, 0, stream>>>(x, lam, sigma, p, q, xbar);
    pd_phaseB<<<NBLOCK, 256, 0, stream>>>(tau, theta, p, q, x0, xbar,
                                          (s == TSTEPS - 1) ? out : nullptr);
  }
}